// GCN_layer_30150670418609
// MI455X (gfx1250) — compile-verified
//
#include <hip/hip_runtime.h>
#include <hip/hip_bf16.h>

typedef __attribute__((ext_vector_type(2))) float v2f;
typedef __attribute__((ext_vector_type(8))) float v8f;

#define FT 128  // FT_IN == FT_OT == 128

// ---------------------------------------------------------------------------
// Kernel 1: zero the h accumulator (workspace is poisoned, must re-init).
// ---------------------------------------------------------------------------
__global__ __launch_bounds__(256) void gcn_zero_ws(float4* __restrict__ p, long n4) {
    long i = (long)blockIdx.x * blockDim.x + threadIdx.x;
    if (i < n4) p[i] = make_float4(0.f, 0.f, 0.f, 0.f);
}

// ---------------------------------------------------------------------------
// Kernel 2: edge-parallel SpMM scatter.
//   One wave32 per edge. Lane l handles features [4l, 4l+4): float4 gather
//   from node_fts[col], scale by w, 4x global_atomic_add_f32 into h[row].
//   h (51.2 MB) is L2-resident on MI455X (192 MB L2), so atomics stay on-chip.
// ---------------------------------------------------------------------------
__global__ __launch_bounds__(256) void gcn_spmm_scatter(
        const float* __restrict__ node_fts,
        const int*   __restrict__ edge_row,
        const int*   __restrict__ edge_col,
        const float* __restrict__ edge_w,
        float*       __restrict__ h,
        int n_edges) {
    const int e    = (int)(((long)blockIdx.x * blockDim.x + threadIdx.x) >> 5);
    const int lane = threadIdx.x & 31;
    if (e >= n_edges) return;

    const int   r = edge_row[e];   // wave-uniform -> scalarized by compiler
    const int   c = edge_col[e];
    const float w = edge_w[e];

    const float4 v = *(const float4*)(node_fts + (long)c * FT + lane * 4);
    float* dst = h + (long)r * FT + lane * 4;
    atomicAdd(dst + 0, w * v.x);
    atomicAdd(dst + 1, w * v.y);
    atomicAdd(dst + 2, w * v.z);
    atomicAdd(dst + 3, w * v.w);
}

// ---------------------------------------------------------------------------
// Kernel 3: dense out = relu(h @ W^T + b) via V_WMMA_F32_16X16X4_F32.
//   One wave per 16x16 output tile. Block = 256 threads = 8 waves, each wave
//   takes one of the 8 N-tiles (128/16); blockIdx.x is the M-tile.
//
//   A (16x4 f32, MxK) lane layout: lanes 0-15 -> {K=0,K=1}, lanes 16-31 ->
//   {K=2,K=3}, M = lane&15  => contiguous float2 load from h row.
//   B (4x16 f32, KxN): B[k][n] = W[n][k] (row-major W), same half-wave K
//   split, N = lane&15      => contiguous float2 load from W row.
//   C/D (16x16 f32): VGPR r -> row (r + 8*(lane>=16)), col = lane&15.
// ---------------------------------------------------------------------------
__global__ __launch_bounds__(256) void gcn_gemm_bias_relu(
        const float* __restrict__ h,
        const float* __restrict__ W,
        const float* __restrict__ bias,
        float*       __restrict__ out,
        int n_nodes) {
    const int lane = threadIdx.x & 31;
    const int wv   = threadIdx.x >> 5;   // 0..7 -> N tile
    const int half = lane >> 4;          // 0 or 1
    const int lrow = lane & 15;

    const int m0 = blockIdx.x * 16;
    const int n0 = wv * 16;

    // Clamp A-row index so EXEC stays all-ones for WMMA even on a ragged tail.
    int arow = m0 + lrow;
    if (arow > n_nodes - 1) arow = n_nodes - 1;

    const float* __restrict__ ha = h + (long)arow * FT + 2 * half;
    const float* __restrict__ wb = W + (long)(n0 + lrow) * FT + 2 * half;

    v8f c = {};
#pragma unroll
    for (int k = 0; k < FT; k += 4) {
        v2f a = *(const v2f*)(ha + k);
        v2f b = *(const v2f*)(wb + k);
        // emits v_wmma_f32_16x16x4_f32 (8-arg f32 pattern, probe-confirmed family)
        c = __builtin_amdgcn_wmma_f32_16x16x4_f32(
                /*neg_a=*/false, a, /*neg_b=*/false, b,
                /*c_mod=*/(short)0, c, /*reuse_a=*/false, /*reuse_b=*/false);
    }

    const float bv = bias[n0 + lrow];
    const int row_base = m0 + half * 8;
#pragma unroll
    for (int r = 0; r < 8; ++r) {
        const int row = row_base + r;
        if (row < n_nodes) {
            float v = c[r] + bv;
            out[(long)row * FT + n0 + lrow] = v > 0.f ? v : 0.f;
        }
    }
}

// ---------------------------------------------------------------------------
// Launcher: zero h -> SpMM scatter -> WMMA GEMM+ReLU, all on `stream`.
//   d_in: [0] node_fts f32 [N,128]  [1] edge_row i32 [E]  [2] edge_col i32 [E]
//         [3] edge_weight f32 [E]   [4] W f32 [128,128]   [5] b f32 [128]
//         [6] n_nodes (scalar)      d_ws: h accumulator (N*128 f32 = 51.2 MB)
// ---------------------------------------------------------------------------
extern "C" void kernel_launch(void* const* d_in, const int* in_sizes, int n_in,
                              void* d_out, int out_size, void* d_ws, size_t ws_size,
                              hipStream_t stream) {
    const float* node_fts = (const float*)d_in[0];
    const int*   edge_row = (const int*)d_in[1];
    const int*   edge_col = (const int*)d_in[2];
    const float* edge_w   = (const float*)d_in[3];
    const float* W        = (const float*)d_in[4];
    const float* bias     = (const float*)d_in[5];

    const int n_nodes = in_sizes[0] / FT;   // 100000
    const int n_edges = in_sizes[1];        // 1.6M

    float* h   = (float*)d_ws;
    float* out = (float*)d_out;

    // 1) h = 0
    {
        const long n4 = (long)n_nodes * FT / 4;
        const int blocks = (int)((n4 + 255) / 256);
        gcn_zero_ws<<<blocks, 256, 0, stream>>>((float4*)h, n4);
    }

    // 2) scatter-accumulate messages (one wave per edge)
    {
        const long threads = (long)n_edges * 32;
        const int blocks = (int)((threads + 255) / 256);
        gcn_spmm_scatter<<<blocks, 256, 0, stream>>>(node_fts, edge_row, edge_col,
                                                     edge_w, h, n_edges);
    }

    // 3) out = relu(h @ W^T + b)  (one wave per 16x16 tile, 8 N-tiles/block)
    {
        const int m_tiles = (n_nodes + 15) / 16;   // 6250
        gcn_gemm_bias_relu<<<m_tiles, 256, 0, stream>>>(h, W, bias, out, n_nodes);
    }
}